// ContextCrossAttention_47682726920728
// MI455X (gfx1250) — compile-verified
//
#include <hip/hip_runtime.h>

// Sizes fixed by the reference
#define BB 8
#define CC 512
#define DD 512
#define HW 16384            // 128*128
#define SCALE 0.044194173824159216f   // 512^-0.5

typedef float v2f __attribute__((ext_vector_type(2)));
typedef float v8f __attribute__((ext_vector_type(8)));

// ---------------------------------------------------------------------------
// Small GEMM on the matrix pipe: out(8 x N) = A8(8 x K) @ op(W) + bias
//   TRANSB=1 : W is (N x K) row-major, B[k][n] = W[n*K + k]   (W^T)
//   TRANSB=0 : W is (K x N) row-major, B[k][n] = W[k*N + n]
// One wave (32 threads) per 16-column tile; rows 8..15 of A are zero padding,
// realized as unconditional clamped loads * lane mask (no EXEC divergence —
// WMMA requires EXEC all-ones). K stepped by 4, unrolled x4.
// ---------------------------------------------------------------------------
template <int TRANSB>
__global__ __launch_bounds__(32)
void gemm8_wmma(const float* __restrict__ A8, const float* __restrict__ W,
                const float* __restrict__ bias, float* __restrict__ out,
                int K, int N)
{
    const int lane  = threadIdx.x;        // 0..31
    const int m     = lane & 15;          // A row / C-D column selector
    const int half  = lane >> 4;          // 0: K {0,1}, 1: K {2,3}
    const int n     = blockIdx.x * 16 + m;
    const int mrow  = m & 7;              // clamped row for the A load
    const float amask = (m < 8) ? 1.0f : 0.0f;

    v8f acc = {};
    #pragma unroll 4
    for (int k0 = 0; k0 < K; k0 += 4) {
        const int ka = k0 + half * 2;
        v2f a, b;
        // A fragment (16x4): lane = row m, two consecutive K values.
        // Unconditional load from a clamped row, zeroed by mask for rows 8..15.
        a.x = A8[mrow * K + ka]     * amask;
        a.y = A8[mrow * K + ka + 1] * amask;
        // B fragment (4x16): lane = column n, two consecutive K values
        if (TRANSB) {
            b.x = W[n * K + ka];
            b.y = W[n * K + ka + 1];
        } else {
            b.x = W[(size_t)(ka)     * N + n];
            b.y = W[(size_t)(ka + 1) * N + n];
        }
        acc = __builtin_amdgcn_wmma_f32_16x16x4_f32(
                  false, a, false, b, (short)0, acc, false, false);
    }

    const float bv = bias ? bias[n] : 0.0f;
    // D layout: lanes 0-15 hold rows 0-7 in acc[0..7]; lanes 16-31 rows 8-15.
    if (half == 0) {
        #pragma unroll
        for (int j = 0; j < 8; ++j)
            out[j * N + n] = acc[j] + bv;
    }
}

// ---------------------------------------------------------------------------
// qkb[b] = q[b] . bk
// ---------------------------------------------------------------------------
__global__ __launch_bounds__(256)
void qdotbk_kernel(const float* __restrict__ q, const float* __restrict__ bk,
                   float* __restrict__ qkb)
{
    __shared__ float s[256];
    const int b = blockIdx.x;
    float acc = 0.0f;
    for (int c = threadIdx.x; c < CC; c += 256)
        acc += q[b * CC + c] * bk[c];
    s[threadIdx.x] = acc;
    __syncthreads();
    for (int off = 128; off > 0; off >>= 1) {
        if (threadIdx.x < off) s[threadIdx.x] += s[threadIdx.x + off];
        __syncthreads();
    }
    if (threadIdx.x == 0) qkb[b] = s[0];
}

// ---------------------------------------------------------------------------
// logits[b,hw] = SCALE * (sum_c qk[b,c] * x[b,c,hw] + qkb[b])
// Streaming pass #1 over x (coalesced along hw), qk staged in LDS.
// ---------------------------------------------------------------------------
__global__ __launch_bounds__(256)
void logits_kernel(const float* __restrict__ x, const float* __restrict__ qk,
                   const float* __restrict__ qkb, float* __restrict__ logits)
{
    __shared__ float qs[CC];
    const int b  = blockIdx.y;
    const int hw = blockIdx.x * 256 + threadIdx.x;
    for (int c = threadIdx.x; c < CC; c += 256) qs[c] = qk[b * CC + c];
    __syncthreads();

    const float* xb = x + (size_t)b * CC * HW + hw;
    float acc = 0.0f;
    #pragma unroll 8
    for (int c = 0; c < CC; ++c)
        acc += qs[c] * xb[(size_t)c * HW];
    logits[b * HW + hw] = (acc + qkb[b]) * SCALE;
}

// ---------------------------------------------------------------------------
// In-place softmax over the HW axis, one block per batch.
// Phase 1: max-reduce; Phase 2: sum of exp; Phase 3: normalize in place.
// ---------------------------------------------------------------------------
__global__ __launch_bounds__(1024)
void softmax_kernel(float* __restrict__ logits)
{
    __shared__ float s[1024];
    float* l = logits + (size_t)blockIdx.x * HW;
    const int t = threadIdx.x;

    float lmax = -3.402823466e+38f;
    for (int i = t; i < HW; i += 1024) lmax = fmaxf(lmax, l[i]);
    s[t] = lmax;
    __syncthreads();
    for (int off = 512; off > 0; off >>= 1) {
        if (t < off) s[t] = fmaxf(s[t], s[t + off]);
        __syncthreads();
    }
    const float m = s[0];
    __syncthreads();

    float lsum = 0.0f;
    for (int i = t; i < HW; i += 1024) lsum += __expf(l[i] - m);
    s[t] = lsum;
    __syncthreads();
    for (int off = 512; off > 0; off >>= 1) {
        if (t < off) s[t] += s[t + off];
        __syncthreads();
    }
    const float inv = 1.0f / s[0];
    __syncthreads();

    for (int i = t; i < HW; i += 1024)
        l[i] = __expf(l[i] - m) * inv;
}

// ---------------------------------------------------------------------------
// pooled[b,c] = sum_hw attn[b,hw] * x[b,c,hw]   (streaming pass #2 over x)
// One block per (c,b); float4 loads; block reduction.
// ---------------------------------------------------------------------------
__global__ __launch_bounds__(256)
void pooled_kernel(const float* __restrict__ x, const float* __restrict__ attn,
                   float* __restrict__ pooled)
{
    __shared__ float s[256];
    const int c = blockIdx.x;
    const int b = blockIdx.y;
    const float4* xb = (const float4*)(x + ((size_t)(b * CC + c)) * HW);
    const float4* ab = (const float4*)(attn + (size_t)b * HW);

    float acc = 0.0f;
    #pragma unroll 4
    for (int i = threadIdx.x; i < HW / 4; i += 256) {
        float4 xv = xb[i];
        float4 av = ab[i];
        acc += av.x * xv.x + av.y * xv.y + av.z * xv.z + av.w * xv.w;
    }
    s[threadIdx.x] = acc;
    __syncthreads();
    for (int off = 128; off > 0; off >>= 1) {
        if (threadIdx.x < off) s[threadIdx.x] += s[threadIdx.x + off];
        __syncthreads();
    }
    if (threadIdx.x == 0) pooled[b * CC + c] = s[0];
}

// ---------------------------------------------------------------------------
// out[b,c,hw] = x[b,c,hw] * gate[b,c]   (streaming pass #3, float4)
// ---------------------------------------------------------------------------
__global__ __launch_bounds__(256)
void scale_kernel(const float* __restrict__ x, const float* __restrict__ gate,
                  float* __restrict__ out)
{
    const size_t i = (size_t)blockIdx.x * 256 + threadIdx.x;   // float4 index
    float4 v = ((const float4*)x)[i];
    const float g = gate[i >> 12];                             // 4096 float4 per (b,c)
    float4 r = {v.x * g, v.y * g, v.z * g, v.w * g};
    ((float4*)out)[i] = r;
}

// ---------------------------------------------------------------------------
extern "C" void kernel_launch(void* const* d_in, const int* in_sizes, int n_in,
                              void* d_out, int out_size, void* d_ws, size_t ws_size,
                              hipStream_t stream)
{
    const float* x       = (const float*)d_in[0];   // (B,C,H,W)
    const float* context = (const float*)d_in[1];   // (B,D)
    const float* Wq      = (const float*)d_in[2];   // (C,D)
    const float* bq      = (const float*)d_in[3];   // (C)
    const float* Wk      = (const float*)d_in[4];   // (C,C)
    const float* bk      = (const float*)d_in[5];   // (C)
    const float* Wv      = (const float*)d_in[6];   // (C,C)
    const float* bv      = (const float*)d_in[7];   // (C)
    float* out = (float*)d_out;

    float* ws     = (float*)d_ws;
    float* q      = ws;              // 4096
    float* qk     = ws + 4096;       // 4096
    float* pooled = ws + 8192;       // 4096
    float* gate   = ws + 12288;      // 4096
    float* qkb    = ws + 16384;      // 8 (padded)
    float* logits = ws + 16640;      // B*HW = 131072 (reused as attn in place)

    // q = context @ Wq^T + bq           (WMMA, W^T path)
    gemm8_wmma<1><<<CC / 16, 32, 0, stream>>>(context, Wq, bq, q, DD, CC);
    // qk = q @ Wk                       (WMMA, plain path)
    gemm8_wmma<0><<<CC / 16, 32, 0, stream>>>(q, Wk, nullptr, qk, CC, CC);
    // qkb[b] = q[b] . bk
    qdotbk_kernel<<<BB, 256, 0, stream>>>(q, bk, qkb);
    // logits over x (pass 1)
    logits_kernel<<<dim3(HW / 256, BB), 256, 0, stream>>>(x, qk, qkb, logits);
    // softmax in place -> attn
    softmax_kernel<<<BB, 1024, 0, stream>>>(logits);
    // attention pooling over x (pass 2)
    pooled_kernel<<<dim3(CC, BB), 256, 0, stream>>>(x, logits, pooled);
    // gate = pooled @ Wv^T + bv         (WMMA, W^T path)
    gemm8_wmma<1><<<CC / 16, 32, 0, stream>>>(pooled, Wv, bv, gate, CC, CC);
    // out = x * gate (pass 3)
    scale_kernel<<<(BB * CC * HW / 4) / 256, 256, 0, stream>>>(x, gate, out);
}